// LSSLfDiagSSM_42743514529995
// MI455X (gfx1250) — compile-verified
//
#include <hip/hip_runtime.h>

// Problem dims (fixed by reference)
#define Hn 128
#define Ln 2048
#define NBn 64
#define CT 64            // chunk length
#define NC (Ln / CT)     // 32 chunks

typedef __attribute__((ext_vector_type(2))) float v2f;
typedef __attribute__((ext_vector_type(8))) float v8f;

// complex integer power by binary exponentiation (exact-ish, ~6 cmuls)
__device__ __forceinline__ void cpow(float br, float bi, int e, float& pr, float& pi) {
    pr = 1.f; pi = 0.f;
    while (e) {
        if (e & 1) { float nr = pr * br - pi * bi; pi = pr * bi + pi * br; pr = nr; }
        float nb = br * br - bi * bi; bi = 2.f * br * bi; br = nb;
        e >>= 1;
    }
}

// ---------------------------------------------------------------------------
// K0: per-head precompute.
//  Mt[h][t][k]  = (k<=t) ? 2Re(sum_n C_n Lam_n^(t-k) B_n) : 0     (64x64 Toeplitz)
//  A2[h][t][:]  = [ 2Re(C Lam^(t+1)) , -2Im(C Lam^(t+1)) ]        (64x128, real GEMM A)
//  L64[h][n]    = Lam_n^64                                         (inter-chunk decay)
// ---------------------------------------------------------------------------
__global__ void k_precompute(const float* Lre, const float* Lim,
                             const float* Bre, const float* Bim,
                             const float* Cre, const float* Cim,
                             float* Mt, float* A2, float* L64r, float* L64i) {
    int h = blockIdx.x, t = threadIdx.x;   // 0..63
    __shared__ float G[CT];
    const float *lr = Lre + h * NBn, *li = Lim + h * NBn;
    const float *br = Bre + h * NBn, *bi = Bim + h * NBn;
    const float *cr = Cre + h * NBn, *ci = Cim + h * NBn;

    float g = 0.f;
    for (int n = 0; n < NBn; n++) {
        float pr, pi; cpow(lr[n], li[n], t, pr, pi);
        float wr = cr[n] * br[n] - ci[n] * bi[n];
        float wi = cr[n] * bi[n] + ci[n] * br[n];
        g += 2.f * (wr * pr - wi * pi);
    }
    G[t] = g;

    float* a2 = A2 + ((size_t)h * CT + t) * (2 * NBn);
    for (int n = 0; n < NBn; n++) {
        float pr, pi; cpow(lr[n], li[n], t + 1, pr, pi);
        float wr = cr[n] * pr - ci[n] * pi;
        float wi = cr[n] * pi + ci[n] * pr;
        a2[n] = 2.f * wr;
        a2[NBn + n] = -2.f * wi;
    }
    { float pr, pi; cpow(lr[t], li[t], CT, pr, pi); L64r[h * NBn + t] = pr; L64i[h * NBn + t] = pi; }

    __syncthreads();
    float* mt = Mt + ((size_t)h * CT + t) * CT;
    for (int k = 0; k < CT; k++) mt[k] = (k <= t) ? G[t - k] : 0.f;
}

// ---------------------------------------------------------------------------
// K2: per-(head,chunk) exact (reset-aware) local scan from zero state.
// Produces chunk summaries b = x_local[T-1], anyreset flag, first-reset index.
// One wave per chunk: lane owns states n and n+32.
// ---------------------------------------------------------------------------
__global__ void k_chunk(const float* u, const int* d,
                        const float* Lre, const float* Lim,
                        const float* Bre, const float* Bim,
                        float* bsr, float* bsi, int* afl, int* frst) {
    int blk = blockIdx.x; int h = blk / NC, c = blk % NC; int lane = threadIdx.x;
    int n0 = lane, n1 = lane + 32;
    float l0r = Lre[h * NBn + n0], l0i = Lim[h * NBn + n0];
    float l1r = Lre[h * NBn + n1], l1i = Lim[h * NBn + n1];
    float b0r = Bre[h * NBn + n0], b0i = Bim[h * NBn + n0];
    float b1r = Bre[h * NBn + n1], b1i = Bim[h * NBn + n1];
    float x0r = 0.f, x0i = 0.f, x1r = 0.f, x1i = 0.f;
    int fr = CT;
    const float* uh = u + (size_t)h * Ln + c * CT;
    const int* dh = d + c * CT;
    for (int t = 0; t < CT; t++) {
        float uv = uh[t];
        bool rs = (dh[t] != 0);
        if (rs && t < fr) fr = t;
        if (rs) {
            x0r = b0r * uv; x0i = b0i * uv; x1r = b1r * uv; x1i = b1i * uv;
        } else {
            float nr;
            nr = l0r * x0r - l0i * x0i + b0r * uv; x0i = l0r * x0i + l0i * x0r + b0i * uv; x0r = nr;
            nr = l1r * x1r - l1i * x1i + b1r * uv; x1i = l1r * x1i + l1i * x1r + b1i * uv; x1r = nr;
        }
    }
    size_t base = ((size_t)h * NC + c) * NBn;
    bsr[base + n0] = x0r; bsi[base + n0] = x0i;
    bsr[base + n1] = x1r; bsi[base + n1] = x1i;
    if (lane == 0) { afl[h * NC + c] = (fr < CT); frst[h * NC + c] = fr; }
}

// ---------------------------------------------------------------------------
// K3: per-head inter-chunk scan (32 sequential steps). Emits x_start per chunk
// and the final state (second output, complex64 interleaved).
// ---------------------------------------------------------------------------
__global__ void k_inter(const float* x0r_in, const float* x0i_in,
                        const float* L64r, const float* L64i,
                        const float* bsr, const float* bsi, const int* afl,
                        float* xsr, float* xsi, float* out) {
    int h = blockIdx.x, lane = threadIdx.x;
    for (int half = 0; half < 2; half++) {
        int n = lane + 32 * half;
        float xr = x0r_in[h * NBn + n], xi = x0i_in[h * NBn + n];
        float ar = L64r[h * NBn + n], ai = L64i[h * NBn + n];
        for (int c = 0; c < NC; c++) {
            size_t base = ((size_t)h * NC + c) * NBn;
            xsr[base + n] = xr; xsi[base + n] = xi;
            if (afl[h * NC + c]) {       // a reset inside chunk kills decay product
                xr = bsr[base + n]; xi = bsi[base + n];
            } else {
                float nr = ar * xr - ai * xi + bsr[base + n];
                xi = ar * xi + ai * xr + bsi[base + n];
                xr = nr;
            }
        }
        out[(size_t)Hn * Ln + 2 * ((size_t)h * NBn + n) + 0] = xr;
        out[(size_t)Hn * Ln + 2 * ((size_t)h * NBn + n) + 1] = xi;
    }
}

// ---------------------------------------------------------------------------
// K4: per-head fused WMMA GEMMs (fp32, V_WMMA_F32_16X16X4_F32):
//   Y[t,c] = sum_k Mt[t,k]*u[c*64+k]  +  sum_kk A2[t,kk]*X2[kk,c]  + D*u
// 256 threads = 8 waves: 4 M-tiles (t) x 2 N-tiles (chunk c).
// A 16x4 layout: VGPR0 = {K0 | K2}, VGPR1 = {K1 | K3} split at lane 16.
// B 4x16 layout: VGPR0 = rows {0 | 2}, VGPR1 = rows {1 | 3}.
// ---------------------------------------------------------------------------
__global__ void k_gemm(const float* u, const float* Mt, const float* A2,
                       const float* xsr, const float* xsi,
                       const float* Dv, float* out) {
    int h = blockIdx.x;
    int wave = threadIdx.x >> 5, lane = threadIdx.x & 31;
    int mt = wave & 3, nt = wave >> 2;
    int mrow = lane & 15;                 // M row (A) / N col (B,D)
    int koff = (lane >> 4) << 1;          // 0 or 2

    v8f acc = {};
    const float* mth = Mt + (size_t)h * CT * CT;
    const float* uh = u + (size_t)h * Ln;

    // GEMM1: Toeplitz intra-chunk filter, K = 64
    for (int k0 = 0; k0 < CT; k0 += 4) {
        v2f a, b;
        a.x = mth[(mt * 16 + mrow) * CT + k0 + koff];
        a.y = mth[(mt * 16 + mrow) * CT + k0 + koff + 1];
        b.x = uh[(nt * 16 + mrow) * CT + k0 + koff];
        b.y = uh[(nt * 16 + mrow) * CT + k0 + koff + 1];
        acc = __builtin_amdgcn_wmma_f32_16x16x4_f32(false, a, false, b,
                                                    (short)0, acc, false, false);
    }
    // GEMM2: state injection, K = 128 (complex expanded, signs folded into A2)
    const float* a2h = A2 + (size_t)h * CT * (2 * NBn);
    {
        int c = nt * 16 + mrow;
        size_t xb = ((size_t)h * NC + c) * NBn;
        for (int k0 = 0; k0 < 2 * NBn; k0 += 4) {
            v2f a, b;
            a.x = a2h[(mt * 16 + mrow) * (2 * NBn) + k0 + koff];
            a.y = a2h[(mt * 16 + mrow) * (2 * NBn) + k0 + koff + 1];
            int kk = k0 + koff;
            b.x = (kk < NBn) ? xsr[xb + kk] : xsi[xb + kk - NBn];
            b.y = (kk + 1 < NBn) ? xsr[xb + kk + 1] : xsi[xb + kk + 1 - NBn];
            acc = __builtin_amdgcn_wmma_f32_16x16x4_f32(false, a, false, b,
                                                        (short)0, acc, false, false);
        }
    }
    // Store: D layout — VGPR r holds M=r (lanes<16) / M=r+8 (lanes>=16), N=lane&15
    float dv = Dv[h];
    int c = nt * 16 + (lane & 15);
    int mb = (lane >> 4) << 3;
    for (int r = 0; r < 8; r++) {
        int t = mt * 16 + r + mb;
        size_t l = (size_t)c * CT + t;
        out[(size_t)h * Ln + l] = acc[r] + dv * uh[l];
    }
}

// ---------------------------------------------------------------------------
// K5: sparse reset fix-up. For chunks containing a reset, true x_t for
// t >= firstReset equals the reset-aware local scan (x_start influence dead),
// so recompute y exactly there and overwrite.
// ---------------------------------------------------------------------------
__global__ void k_fix(const float* u, const int* d,
                      const float* Lre, const float* Lim,
                      const float* Bre, const float* Bim,
                      const float* Cre, const float* Cim,
                      const float* Dv, const int* frst, float* out) {
    int blk = blockIdx.x; int h = blk / NC, c = blk % NC; int lane = threadIdx.x;
    int fr = frst[h * NC + c];
    if (fr >= CT) return;
    int n0 = lane, n1 = lane + 32;
    float l0r = Lre[h * NBn + n0], l0i = Lim[h * NBn + n0];
    float l1r = Lre[h * NBn + n1], l1i = Lim[h * NBn + n1];
    float b0r = Bre[h * NBn + n0], b0i = Bim[h * NBn + n0];
    float b1r = Bre[h * NBn + n1], b1i = Bim[h * NBn + n1];
    float c0r = Cre[h * NBn + n0], c0i = Cim[h * NBn + n0];
    float c1r = Cre[h * NBn + n1], c1i = Cim[h * NBn + n1];
    float dv = Dv[h];
    float x0r = 0.f, x0i = 0.f, x1r = 0.f, x1i = 0.f;
    const float* uh = u + (size_t)h * Ln + c * CT;
    const int* dh = d + c * CT;
    for (int t = 0; t < CT; t++) {
        float uv = uh[t];
        bool rs = (dh[t] != 0);
        if (rs) {
            x0r = b0r * uv; x0i = b0i * uv; x1r = b1r * uv; x1i = b1i * uv;
        } else {
            float nr;
            nr = l0r * x0r - l0i * x0i + b0r * uv; x0i = l0r * x0i + l0i * x0r + b0i * uv; x0r = nr;
            nr = l1r * x1r - l1i * x1i + b1r * uv; x1i = l1r * x1i + l1i * x1r + b1i * uv; x1r = nr;
        }
        if (t >= fr) {
            float p = c0r * x0r - c0i * x0i + c1r * x1r - c1i * x1i;
            for (int off = 16; off; off >>= 1) p += __shfl_xor(p, off, 32);
            if (lane == 0) out[(size_t)h * Ln + (size_t)c * CT + t] = 2.f * p + dv * uv;
        }
    }
}

// ---------------------------------------------------------------------------
extern "C" void kernel_launch(void* const* d_in, const int* in_sizes, int n_in,
                              void* d_out, int out_size, void* d_ws, size_t ws_size,
                              hipStream_t stream) {
    const float* u   = (const float*)d_in[0];
    const float* xre = (const float*)d_in[1];
    const float* xim = (const float*)d_in[2];
    const float* Lre = (const float*)d_in[3];
    const float* Lim = (const float*)d_in[4];
    const float* Bre = (const float*)d_in[5];
    const float* Bim = (const float*)d_in[6];
    const float* Cre = (const float*)d_in[7];
    const float* Cim = (const float*)d_in[8];
    const float* Dv  = (const float*)d_in[9];
    const int*   dd  = (const int*)d_in[10];
    float* out = (float*)d_out;

    // workspace layout (floats)
    float* w = (float*)d_ws;
    size_t o = 0;
    float* Mt   = w + o; o += (size_t)Hn * CT * CT;        // 524288
    float* A2   = w + o; o += (size_t)Hn * CT * 2 * NBn;   // 1048576
    float* L64r = w + o; o += (size_t)Hn * NBn;
    float* L64i = w + o; o += (size_t)Hn * NBn;
    float* bsr  = w + o; o += (size_t)Hn * NC * NBn;
    float* bsi  = w + o; o += (size_t)Hn * NC * NBn;
    float* xsr  = w + o; o += (size_t)Hn * NC * NBn;
    float* xsi  = w + o; o += (size_t)Hn * NC * NBn;
    int* afl  = (int*)(w + o); o += (size_t)Hn * NC;
    int* frst = (int*)(w + o); o += (size_t)Hn * NC;

    k_precompute<<<Hn, CT, 0, stream>>>(Lre, Lim, Bre, Bim, Cre, Cim, Mt, A2, L64r, L64i);
    k_chunk<<<Hn * NC, 32, 0, stream>>>(u, dd, Lre, Lim, Bre, Bim, bsr, bsi, afl, frst);
    k_inter<<<Hn, 32, 0, stream>>>(xre, xim, L64r, L64i, bsr, bsi, afl, xsr, xsi, out);
    k_gemm<<<Hn, 256, 0, stream>>>(u, Mt, A2, xsr, xsi, Dv, out);
    k_fix<<<Hn * NC, 32, 0, stream>>>(u, dd, Lre, Lim, Bre, Bim, Cre, Cim, Dv, frst, out);
}